// PagedAttention_11725260718468
// MI455X (gfx1250) — compile-verified
//
#include <hip/hip_runtime.h>

typedef __attribute__((ext_vector_type(2))) float v2f;
typedef __attribute__((ext_vector_type(8))) float v8f;

#define B_    32
#define H_    32
#define D_    128
#define HKV_  8
#define G_    4
#define BS_   16
#define MB_   256
#define NBLK_ (B_ * MB_)
#define NWAVES 8
#define KSTRIDE 132   // 128 + 4 pad -> conflict-free column reads
#define PSTRIDE 17    // 16 + 1 pad
#define NEGINF (-3.0e38f)

__global__ __launch_bounds__(256) void paged_attn_decode_kernel(
    const float* __restrict__ query,
    const float* __restrict__ key_new,
    const float* __restrict__ value_new,
    const float* __restrict__ kv_cache,
    const int*   __restrict__ block_tables,
    const int*   __restrict__ seq_lens,
    float* __restrict__ out)
{
    __shared__ float sK[NWAVES][BS_ * KSTRIDE];
    __shared__ float sV[NWAVES][BS_ * KSTRIDE];
    __shared__ float sP[NWAVES][BS_ * PSTRIDE];
    __shared__ float sQ[G_ * KSTRIDE];
    __shared__ float sOut[NWAVES * G_ * D_];
    __shared__ float sM[NWAVES * G_];
    __shared__ float sL[NWAVES * G_];

    const int wg   = blockIdx.x;        // b * HKV + kvh
    const int b    = wg / HKV_;
    const int kvh  = wg % HKV_;
    const int tid  = threadIdx.x;
    const int wave = tid >> 5;
    const int lane = tid & 31;
    const int lo   = lane & 15;
    const int hi   = lane >> 4;

    // ---- stage Q (G x D) into LDS (query head h = g*HKV + kvh) ----
    for (int i = tid; i < G_ * D_; i += 256) {
        int g = i >> 7, d = i & 127;
        sQ[g * KSTRIDE + d] =
            query[(size_t)b * (H_ * D_) + (size_t)(g * HKV_ + kvh) * D_ + d];
    }
    __syncthreads();

    const int   seq_len = seq_lens[b];
    const int   n_tiles = (seq_len + BS_ - 1) / BS_;
    const float scale   = 0.08838834764831845f;  // 1/sqrt(128)

    // ---- hoist Q^T fragments (scale folded in): B-matrix (4 x 16) per d-chunk ----
    // B layout (32-bit 4x16): vgpr0 = K rows {0,2}, vgpr1 = rows {1,3};
    // lanes 0-15 -> rows 0/1, lanes 16-31 -> rows 2/3; N (col) = lane&15 = g.
    v2f qf[32];
    {
        const bool gv = (lo < G_);
        #pragma unroll
        for (int c = 0; c < 32; ++c) {
            int dd = 4 * c + 2 * hi;
            qf[c].x = gv ? sQ[lo * KSTRIDE + dd]     * scale : 0.0f;
            qf[c].y = gv ? sQ[lo * KSTRIDE + dd + 1] * scale : 0.0f;
        }
    }

    // out^T accumulators: 8 chunks of (16 d-rows x 16 g-cols); lane's N = g = lo
    v8f  acc[8];
    #pragma unroll
    for (int i = 0; i < 8; ++i)
        #pragma unroll
        for (int r = 0; r < 8; ++r) acc[i][r] = 0.0f;

    float m_run = NEGINF;
    float l_run = 0.0f;

    float* myK = sK[wave];
    float* myV = sV[wave];
    float* myP = sP[wave];

    const size_t VHALF = (size_t)NBLK_ * BS_ * HKV_ * D_;
    const float* knew  = key_new   + (size_t)b * HKV_ * D_ + (size_t)kvh * D_;
    const float* vnew  = value_new + (size_t)b * HKV_ * D_ + (size_t)kvh * D_;

    for (int t = wave; t < n_tiles; t += NWAVES) {
        const int    blk = block_tables[b * MB_ + t];
        const float* kb  = kv_cache + ((size_t)blk * BS_ * HKV_ + kvh) * D_;
        const float* vb  = kb + VHALF;

        // prefetch next tile owned by this wave (global_prefetch_b8)
        int tn = t + NWAVES;
        if (tn < n_tiles) {
            int blkn = block_tables[b * MB_ + tn];
            const float* kpn = kv_cache + ((size_t)blkn * BS_ * HKV_ + kvh) * D_;
            __builtin_prefetch(kpn + (size_t)lo * HKV_ * D_ + hi * 64, 0, 0);
            __builtin_prefetch(kpn + VHALF + (size_t)lo * HKV_ * D_ + hi * 64, 0, 0);
        }

        // ---- stage K/V tile (16 x 128) to LDS; row == fresh token -> inputs ----
        #pragma unroll 4
        for (int r = 0; r < BS_; ++r) {
            const int    p    = t * BS_ + r;
            const bool   fr   = (p == seq_len - 1);
            const float* ksrc = fr ? knew : (kb + (size_t)r * HKV_ * D_);
            const float* vsrc = fr ? vnew : (vb + (size_t)r * HKV_ * D_);
            float4 k4 = ((const float4*)ksrc)[lane];   // 32 lanes x 16B = full row
            float4 v4 = ((const float4*)vsrc)[lane];
            *(float4*)&myK[r * KSTRIDE + lane * 4] = k4;
            *(float4*)&myV[r * KSTRIDE + lane * 4] = v4;
        }
        asm volatile("s_wait_dscnt 0" ::: "memory");  // cross-lane LDS visibility

        // ---- scores^T = K_tile(16x128) * (Q*scale)^T(128x16) ----
        // 32 f32 WMMAs split over two independent accumulator chains.
        // A (16x4, 32-bit): lane row M = key = lo; vgpr0 K={0,2}, vgpr1 K={1,3}
        v8f sc0, sc1;
        #pragma unroll
        for (int r = 0; r < 8; ++r) { sc0[r] = 0.0f; sc1[r] = 0.0f; }
        #pragma unroll
        for (int c = 0; c < 32; c += 2) {
            int dd0 = 4 * c + 2 * hi;
            v2f a0 = *(const v2f*)&myK[lo * KSTRIDE + dd0];
            sc0 = __builtin_amdgcn_wmma_f32_16x16x4_f32(
                false, a0, false, qf[c], (short)0, sc0, false, false);
            v2f a1 = *(const v2f*)&myK[lo * KSTRIDE + dd0 + 4];
            sc1 = __builtin_amdgcn_wmma_f32_16x16x4_f32(
                false, a1, false, qf[c + 1], (short)0, sc1, false, false);
        }

        // ---- online softmax over this tile ----
        // D layout: vgpr r -> key = r + hi*8, col g = lo
        float sv[8];
        float mloc = NEGINF;
        #pragma unroll
        for (int r = 0; r < 8; ++r) {
            int   p = t * BS_ + r + hi * 8;
            float s = (p < seq_len) ? (sc0[r] + sc1[r]) : NEGINF;
            sv[r] = s;
            mloc  = fmaxf(mloc, s);
        }
        mloc = fmaxf(mloc, __shfl_xor(mloc, 16, 32));
        float m_new = fmaxf(m_run, mloc);
        float alpha = __expf(m_run - m_new);
        float lloc  = 0.0f;
        #pragma unroll
        for (int r = 0; r < 8; ++r) {
            float pe = __expf(sv[r] - m_new);
            myP[(r + hi * 8) * PSTRIDE + lo] = pe;   // P^T[key][g]
            lloc += pe;
        }
        lloc += __shfl_xor(lloc, 16, 32);
        l_run = l_run * alpha + lloc;
        m_run = m_new;
        #pragma unroll
        for (int i = 0; i < 8; ++i)
            #pragma unroll
            for (int r = 0; r < 8; ++r) acc[i][r] *= alpha;  // alpha is per-g (= per-lane)
        asm volatile("s_wait_dscnt 0" ::: "memory");

        // ---- out^T += V^T(128x16) * P^T(16x16): 32 f32 WMMAs, 8 indep chains ----
        #pragma unroll
        for (int j = 0; j < 4; ++j) {
            int srow = 4 * j + 2 * hi;
            v2f bp;                                   // B: P^T rows {srow, srow+1}
            bp.x = myP[srow * PSTRIDE + lo];
            bp.y = myP[(srow + 1) * PSTRIDE + lo];
            #pragma unroll
            for (int i = 0; i < 8; ++i) {
                v2f av;                               // A: V^T, M = d = 16i + lo
                av.x = myV[srow * KSTRIDE + i * 16 + lo];
                av.y = myV[(srow + 1) * KSTRIDE + i * 16 + lo];
                acc[i] = __builtin_amdgcn_wmma_f32_16x16x4_f32(
                    false, av, false, bp, (short)0, acc[i], false, false);
            }
        }
    }

    // ---- cross-wave flash-decoding merge ----
    if (lo < G_ && hi == 0) {
        sM[wave * G_ + lo] = m_run;
        sL[wave * G_ + lo] = l_run;
    }
    if (lo < G_) {
        #pragma unroll
        for (int i = 0; i < 8; ++i)
            #pragma unroll
            for (int r = 0; r < 8; ++r) {
                int d = i * 16 + r + hi * 8;
                sOut[(wave * G_ + lo) * D_ + d] = acc[i][r];
            }
    }
    __syncthreads();

    for (int o = tid; o < G_ * D_; o += 256) {
        int   g = o >> 7, d = o & 127;
        float M = NEGINF;
        for (int w = 0; w < NWAVES; ++w) M = fmaxf(M, sM[w * G_ + g]);
        float Lsum = 0.0f, val = 0.0f;
        for (int w = 0; w < NWAVES; ++w) {
            float e = __expf(sM[w * G_ + g] - M);
            Lsum += e * sL[w * G_ + g];
            val  += e * sOut[(w * G_ + g) * D_ + d];
        }
        out[(size_t)b * (H_ * D_) + (size_t)(g * HKV_ + kvh) * D_ + d] = val / Lsum;
    }
}

extern "C" void kernel_launch(void* const* d_in, const int* in_sizes, int n_in,
                              void* d_out, int out_size, void* d_ws, size_t ws_size,
                              hipStream_t stream) {
    (void)in_sizes; (void)n_in; (void)out_size; (void)d_ws; (void)ws_size;
    const float* query    = (const float*)d_in[0];
    const float* key_new  = (const float*)d_in[1];
    const float* value    = (const float*)d_in[2];
    const float* kv_cache = (const float*)d_in[3];
    const int*   btables  = (const int*)d_in[4];
    // d_in[5] = slot_mapping (folded into the gather), d_in[7] = is_prefill (decode-only ref)
    const int*   seqlens  = (const int*)d_in[6];
    float*       out      = (float*)d_out;

    paged_attn_decode_kernel<<<dim3(B_ * HKV_), dim3(256), 0, stream>>>(
        query, key_new, value, kv_cache, btables, seqlens, out);
}